// YOLOPredictor_57174604644477
// MI455X (gfx1250) — compile-verified
//
#include <hip/hip_runtime.h>
#include <stdint.h>

// YOLOv3 head decode for MI455X (gfx1250).
// Memory-bound transpose+activation: staged through LDS via CDNA5 async
// global->LDS DMA (ASYNCcnt), written back with non-temporal stores.
//
// Layout: in  [B, 3*85, 76, 76]  (attr-major planes, spatial contiguous)
//         out [B, 3*76*76, 85]   (attr innermost)
// Per tile: one (b, anchor, grid-row y) -> 85 x 76 floats.

#define THREADS 256

__global__ __launch_bounds__(THREADS) void yolo_head_decode(
    const float* __restrict__ in, float* __restrict__ out) {
  constexpr int NA    = 3;
  constexpr int ATTRS = 85;           // 5 + 80 classes
  constexpr int GH    = 76;
  constexpr int GW    = 76;
  constexpr int PLANE = GH * GW;      // 5776
  constexpr int TILE  = ATTRS * GW;   // 6460 elements per workgroup
  constexpr int PITCH = GW + 1;       // 77 dwords, odd -> conflict-free banked reads

  __shared__ float tile[ATTRS * PITCH];   // ~26.2 KB

  const int blk = blockIdx.x;
  const int y   = blk % GH;
  const int a   = (blk / GH) % NA;
  const int b   = blk / (GH * NA);

  // anchors in pixels; exp(w)*(anchor/stride)*stride == exp(w)*anchor_px
  const float AW[NA] = {10.0f, 16.0f, 33.0f};
  const float AH[NA] = {13.0f, 30.0f, 23.0f};
  const float aw = AW[a];
  const float ah = AH[a];
  const float fy = (float)y;

  const float* gsrc =
      in + (size_t)((b * NA + a) * ATTRS) * PLANE + (size_t)y * GW;

  // ---- Phase 1: async DMA global -> LDS, raw tile, row-major [attr][col] ----
  // Each attr row is 76 contiguous floats in memory -> coalesced requests.
  for (int i = threadIdx.x; i < TILE; i += THREADS) {
    const int attr = i / GW;
    const int col  = i - attr * GW;
    const float* g = gsrc + (size_t)attr * PLANE + col;
    // low 32 bits of a generic LDS pointer == wave-relative LDS byte offset
    const uint32_t l = (uint32_t)(uintptr_t)(&tile[attr * PITCH + col]);
    asm volatile("global_load_async_to_lds_b32 %0, %1, off"
                 :: "v"(l), "v"(g)
                 : "memory");
  }
  asm volatile("s_wait_asynccnt 0" ::: "memory");
  __syncthreads();

  // ---- Phase 2: transform + transposed write, fully contiguous NT stores ----
  // Thread i emits flat element i of the [76 x 85] output tile.
  float* gdst =
      out + (size_t)((b * NA + a) * PLANE + y * GW) * ATTRS;
  constexpr float LOG2E = 1.44269504088896340736f;
  for (int i = threadIdx.x; i < TILE; i += THREADS) {
    const int col  = i / ATTRS;        // spatial x within this grid row
    const int attr = i - col * ATTRS;  // 0..84
    const float v = tile[attr * PITCH + col];   // stride-77 -> no bank conflicts
    const float e = __builtin_amdgcn_exp2f(v * LOG2E);            // exp(v)
    const float s = __builtin_amdgcn_rcpf(
        1.0f + __builtin_amdgcn_exp2f(v * -LOG2E));               // sigmoid(v)
    float r;
    if (attr == 0)      r = (s + (float)col) * 8.0f;  // (sig+grid_x)*stride_w
    else if (attr == 1) r = (s + fy) * 8.0f;          // (sig+grid_y)*stride_h
    else if (attr == 2) r = e * aw;                   // exp(w)*anchor_w_px
    else if (attr == 3) r = e * ah;                   // exp(h)*anchor_h_px
    else                r = s;                        // conf + 80 classes
    __builtin_nontemporal_store(r, gdst + i);         // streamed once, skip L2 RT
  }
}

extern "C" void kernel_launch(void* const* d_in, const int* in_sizes, int n_in,
                              void* d_out, int out_size, void* d_ws, size_t ws_size,
                              hipStream_t stream) {
  const float* x = (const float*)d_in[0];
  float* out = (float*)d_out;

  constexpr int NA = 3, ATTRS = 85, GH = 76, GW = 76;
  const int batches = in_sizes[0] / (NA * ATTRS * GH * GW);  // 32 for reference shapes

  const dim3 grid(batches * NA * GH);   // one workgroup per (b, anchor, grid row)
  const dim3 block(THREADS);
  yolo_head_decode<<<grid, block, 0, stream>>>(x, out);
}